// Attention_4810363372504
// MI455X (gfx1250) — compile-verified
//
#include <hip/hip_runtime.h>
#include <math.h>

#define BS 4
#define NE 256
#define D  256
#define NEG_INF (-__builtin_inff())

typedef float v2f __attribute__((ext_vector_type(2)));
typedef float v4f __attribute__((ext_vector_type(4)));
typedef float v8f __attribute__((ext_vector_type(8)));

// ---------------------------------------------------------------------------
// WMMA fp32 GEMM (16x16 tile per wave, K stepped by 4 via v_wmma_f32_16x16x4_f32)
//   C[M,N] = A[M,K] * B  (+ bias[n]) (* rowscale[m])
//   MODE 0: B = W^T where W is stored (N,K) row-major (torch Linear weight)
//   MODE 1: B = W stored (K,N) row-major
// Fragment layouts per CDNA5 ISA 7.12.2:
//   A 16x4 : lane&15 = M row; VGPR v holds K = kbase + 2*(lane>=16) + v
//   B 4x16 : lane&15 = N col; VGPR v holds K = kbase + 2*(lane>=16) + v
//   C 16x16: lane&15 = N col; VGPR v holds M = v + 8*(lane>=16)
// ---------------------------------------------------------------------------
template<int MODE, bool BIAS>
__global__ __launch_bounds__(128)
void wmma_gemm(const float* __restrict__ A, const float* __restrict__ W,
               const float* __restrict__ bias, const float* __restrict__ rowscale,
               float* __restrict__ C, int M, int N, int K)
{
    const int lane  = threadIdx.x & 31;
    const int gwave = (blockIdx.x * blockDim.x + threadIdx.x) >> 5;
    const int ntile = N >> 4;
    const int mt = gwave / ntile;
    const int nt = gwave - mt * ntile;
    if (mt >= (M >> 4)) return;           // wave-uniform: EXEC stays all-ones
    const int half = lane >> 4;
    const int l15  = lane & 15;
    const int arow = mt * 16 + l15;
    const int bcol = nt * 16 + l15;

    v8f c = {0.f, 0.f, 0.f, 0.f, 0.f, 0.f, 0.f, 0.f};
    for (int kk = 0; kk < K; kk += 4) {
        const int k0 = kk + 2 * half;
        v2f a, b;
        a[0] = A[(size_t)arow * K + k0];
        a[1] = A[(size_t)arow * K + k0 + 1];
        if (MODE == 0) {
            b[0] = W[(size_t)bcol * K + k0];
            b[1] = W[(size_t)bcol * K + k0 + 1];
        } else {
            b[0] = W[(size_t)(k0    ) * N + bcol];
            b[1] = W[(size_t)(k0 + 1) * N + bcol];
        }
        c = __builtin_amdgcn_wmma_f32_16x16x4_f32(false, a, false, b,
                                                  (short)0, c, false, false);
    }
    const float bv = BIAS ? bias[bcol] : 0.0f;
#pragma unroll
    for (int v = 0; v < 8; ++v) {
        const int m = mt * 16 + v + 8 * half;
        float r = c[v] + bv;
        if (rowscale) r *= rowscale[m];
        C[(size_t)m * N + bcol] = r;
    }
}

// ---------------------------------------------------------------------------
// Bias correction dots: cA[m] = <EQ[m,:], bK>,  c0[m] = <EK[m,:], bQ>
// ---------------------------------------------------------------------------
__global__ __launch_bounds__(256)
void bias_dots(const float* __restrict__ EQ, const float* __restrict__ EK,
               const float* __restrict__ bQ, const float* __restrict__ bK,
               float* __restrict__ cA, float* __restrict__ c0)
{
    const int m    = blockIdx.x * (blockDim.x >> 5) + (threadIdx.x >> 5);
    const int lane = threadIdx.x & 31;
    float pa = 0.f, p0 = 0.f;
#pragma unroll
    for (int v = 0; v < 8; ++v) {
        const int e = lane + 32 * v;
        pa += EQ[(size_t)m * D + e] * bK[e];
        p0 += EK[(size_t)m * D + e] * bQ[e];
    }
#pragma unroll
    for (int o = 16; o >= 1; o >>= 1) {
        pa += __shfl_xor(pa, o, 32);
        p0 += __shfl_xor(p0, o, 32);
    }
    if (lane == 0) { cA[m] = pa; c0[m] = p0; }
}

// element mapping within a wave: e = lane*8 + v  (contiguous 32B per lane)
__device__ __forceinline__ void load8(float r[8], const float* __restrict__ p, int lane)
{
    const v4f* p4 = (const v4f*)p + 2 * lane;
    v4f t0 = p4[0], t1 = p4[1];
    r[0]=t0.x; r[1]=t0.y; r[2]=t0.z; r[3]=t0.w;
    r[4]=t1.x; r[5]=t1.y; r[6]=t1.z; r[7]=t1.w;
}
// Non-temporal variant: R is streamed exactly once -> keep it out of L2 so the
// hot G/E working set (~2 MB) stays resident.
__device__ __forceinline__ void load8_nt(float r[8], const float* __restrict__ p, int lane)
{
    const v4f* p4 = (const v4f*)p + 2 * lane;
    v4f t0 = __builtin_nontemporal_load(p4);
    v4f t1 = __builtin_nontemporal_load(p4 + 1);
    r[0]=t0.x; r[1]=t0.y; r[2]=t0.z; r[3]=t0.w;
    r[4]=t1.x; r[5]=t1.y; r[6]=t1.z; r[7]=t1.w;
}
__device__ __forceinline__ void store8_nt(float* __restrict__ p, const float r[8], int lane)
{
    v4f* p4 = (v4f*)p + 2 * lane;
    v4f t0 = {r[0], r[1], r[2], r[3]};
    v4f t1 = {r[4], r[5], r[6], r[7]};
    __builtin_nontemporal_store(t0, p4);
    __builtin_nontemporal_store(t1, p4 + 1);
}

// ---------------------------------------------------------------------------
// Streaming pass: one workgroup (8 waves) per (b,i) row of R.
// Single read of R; emits R_Z and the alpha-weighted sum S[b,i] (online softmax).
// ---------------------------------------------------------------------------
__global__ __launch_bounds__(256)
void row_kernel(const float* __restrict__ R, const float* __restrict__ E,
                const float* __restrict__ Rmas, const float* __restrict__ Emas,
                const float* __restrict__ F, const float* __restrict__ G,
                const float* __restrict__ EV, const float* __restrict__ cA,
                const float* __restrict__ c0,
                float* __restrict__ RZ, float* __restrict__ S)
{
    const int bi   = blockIdx.x;            // 0 .. BS*NE-1
    const int b    = bi >> 8;
    const int lane = threadIdx.x & 31;
    const int wave = threadIdx.x >> 5;

    const float* Rrow = R    + (size_t)bi * NE * D;
    const float* Gb   = G    + (size_t)b  * NE * D;
    const float* Eb   = E    + (size_t)b  * NE * D;
    const float* Rm   = Rmas + (size_t)bi * NE;
    const float  emI  = Emas[bi];
    const float  caI  = cA[bi];
    const float  c0I  = c0[bi];

    float f[8], gi[8], ev[8];
    load8(f,  F  + (size_t)bi * D, lane);
    load8(gi, G  + (size_t)bi * D, lane);
    load8(ev, EV + (size_t)bi * D, lane);

    float m = NEG_INF, l = 0.f;
    float s[8] = {0.f,0.f,0.f,0.f,0.f,0.f,0.f,0.f};

    for (int j = wave; j < NE; j += 8) {
        float r[8], gj[8], ej[8];
        load8_nt(r, Rrow + (size_t)j * D, lane);  // streamed once: NT
        load8(gj, Gb + (size_t)j * D, lane);      // L2-hot (1 MB total)
        load8(ej, Eb + (size_t)j * D, lane);      // L2-hot
        float pa = 0.f, p0 = 0.f, p1 = 0.f;
#pragma unroll
        for (int v = 0; v < 8; ++v) {
            pa += r[v] * f[v];
            p0 += r[v] * gi[v];
            p1 += r[v] * gj[v];
        }
#pragma unroll
        for (int o = 16; o >= 1; o >>= 1) {      // full-wave sums to all lanes
            pa += __shfl_xor(pa, o, 32);
            p0 += __shfl_xor(p0, o, 32);
            p1 += __shfl_xor(p1, o, 32);
        }
        const float rmj  = Rm[j];
        const float emJ  = Emas[(b << 8) + j];
        const bool amask = (emI * rmj) > 0.f;
        const bool bmask = (emI * emJ * rmj) > 0.f;
        const float la = amask ? (pa + caI) : NEG_INF;
        const float b0 = bmask ? (p0 + c0I) : NEG_INF;
        const float b1 = bmask ? (p1 + c0[(b << 8) + j]) : NEG_INF;
        const float nb0 = 1.f / (1.f + expf(b0 / b1));   // sigmoid(-(b0/b1))
        const float nb1 = 1.f / (1.f + expf(b1 / b0));

        float o8[8];
#pragma unroll
        for (int v = 0; v < 8; ++v) o8[v] = (ev[v] * nb0 + ej[v] * nb1) * rmj;
        store8_nt(RZ + ((size_t)bi * NE + j) * D, o8, lane);  // write-once: NT

        // online softmax accumulation of S = sum_j alpha_j * R[b,i,j,:]
        const float mn = fmaxf(m, la);
        if (mn != NEG_INF) {
            const float sc = (m  == NEG_INF) ? 0.f : expf(m  - mn);
            const float p  = (la == NEG_INF) ? 0.f : expf(la - mn);
            l = l * sc + p;
#pragma unroll
            for (int v = 0; v < 8; ++v) s[v] = s[v] * sc + p * r[v];
            m = mn;
        }
    }

    __shared__ float sm[8], sl[8], ss[8][D];
    if (lane == 0) { sm[wave] = m; sl[wave] = l; }
#pragma unroll
    for (int v = 0; v < 8; ++v) ss[wave][lane * 8 + v] = s[v];
    __syncthreads();

    const int t = threadIdx.x;                    // one output element per thread
    float Mx = sm[0];
#pragma unroll
    for (int w = 1; w < 8; ++w) Mx = fmaxf(Mx, sm[w]);
    float L = 0.f, acc = 0.f;
#pragma unroll
    for (int w = 0; w < 8; ++w) {
        const float sc = (sm[w] == NEG_INF) ? 0.f : expf(sm[w] - Mx);
        L   += sl[w] * sc;
        acc += ss[w][t] * sc;
    }
    S[(size_t)bi * D + t] = acc / L;
}

// ---------------------------------------------------------------------------
extern "C" void kernel_launch(void* const* d_in, const int* in_sizes, int n_in,
                              void* d_out, int out_size, void* d_ws, size_t ws_size,
                              hipStream_t stream)
{
    (void)in_sizes; (void)n_in; (void)out_size; (void)ws_size;
    const float* R    = (const float*)d_in[0];
    const float* E    = (const float*)d_in[1];
    const float* Rmas = (const float*)d_in[2];
    const float* Emas = (const float*)d_in[3];
    const float* WQw  = (const float*)d_in[4];
    const float* WQb  = (const float*)d_in[5];
    const float* WKw  = (const float*)d_in[6];
    const float* WKb  = (const float*)d_in[7];
    const float* WVw  = (const float*)d_in[8];
    const float* WVb  = (const float*)d_in[9];

    float* RZ = (float*)d_out;                              // (BS,NE,NE,D)
    float* EZ = RZ + (size_t)BS * NE * NE * D;              // (BS,NE,D)

    const int M = BS * NE;                                  // 1024
    float* ws = (float*)d_ws;
    float* EQ = ws;               // M*D
    float* EK = EQ + (size_t)M * D;
    float* EV = EK + (size_t)M * D;
    float* F  = EV + (size_t)M * D;
    float* G  = F  + (size_t)M * D;
    float* S  = G  + (size_t)M * D;
    float* cA = S  + (size_t)M * D;   // M
    float* c0 = cA + M;               // M

    const dim3 gblk(128);                                    // 4 waves/block
    const dim3 ggrd(((M / 16) * (D / 16)) / 4);              // 256 blocks, exact

    // Entity projections: E_Q/E_K/E_V = E * W^T + b
    wmma_gemm<0, true ><<<ggrd, gblk, 0, stream>>>(E,  WQw, WQb, nullptr, EQ, M, D, D);
    wmma_gemm<0, true ><<<ggrd, gblk, 0, stream>>>(E,  WKw, WKb, nullptr, EK, M, D, D);
    wmma_gemm<0, true ><<<ggrd, gblk, 0, stream>>>(E,  WVw, WVb, nullptr, EV, M, D, D);
    // Folded projection vectors: F = E_Q * W_K,  G = E_K * W_Q
    wmma_gemm<1, false><<<ggrd, gblk, 0, stream>>>(EQ, WKw, nullptr, nullptr, F, M, D, D);
    wmma_gemm<1, false><<<ggrd, gblk, 0, stream>>>(EK, WQw, nullptr, nullptr, G, M, D, D);
    // Bias correction scalars
    bias_dots<<<dim3(M / 8), dim3(256), 0, stream>>>(EQ, EK, WQb, WKb, cA, c0);
    // Single streaming pass over R: emit R_Z and S
    row_kernel<<<dim3(BS * NE), dim3(256), 0, stream>>>(R, E, Rmas, Emas, F, G, EV,
                                                        cA, c0, RZ, S);
    // E_Z = (S * W_V^T + b_V) * E_mas
    wmma_gemm<0, true ><<<ggrd, gblk, 0, stream>>>(S,  WVw, WVb, Emas, EZ, M, D, D);
}